// Model_11433202942500
// MI455X (gfx1250) — compile-verified
//
#include <hip/hip_runtime.h>
#include <math.h>

#define DIM 64

typedef __attribute__((ext_vector_type(2))) float v2f;
typedef __attribute__((ext_vector_type(8))) float v8f;

// ---------------- utility kernels ----------------

__global__ void fill_kernel(float* __restrict__ p, float v, int n) {
  int i = blockIdx.x * blockDim.x + threadIdx.x;
  if (i < n) p[i] = v;
}

// x[row] = emb[ids[row]]; also seed xsum with the same values. 16 threads/row, float4 each.
__global__ void gather_rows_kernel(const float* __restrict__ emb,
                                   const int* __restrict__ ids,
                                   float* __restrict__ xcur,
                                   float* __restrict__ xsum, int n) {
  int t = blockIdx.x * blockDim.x + threadIdx.x;
  int row = t >> 4;
  if (row >= n) return;
  int c = (t & 15) * 4;
  float4 v = *(const float4*)(emb + (size_t)ids[row] * DIM + c);
  *(float4*)(xcur + (size_t)row * DIM + c) = v;
  *(float4*)(xsum + (size_t)row * DIM + c) = v;
}

__global__ void count_dst_kernel(const int* __restrict__ dst,
                                 float* __restrict__ cnt, int E) {
  int e = blockIdx.x * blockDim.x + threadIdx.x;
  if (e < E) atomicAdd(cnt + dst[e], 1.0f);
}

// acc[dst[e]] += x[src[e]]  (16 threads per edge, float4 chunks, f32 atomics)
__global__ void scatter_rows_kernel(const float* __restrict__ x,
                                    const int* __restrict__ src,
                                    const int* __restrict__ dst,
                                    float* __restrict__ acc, int E) {
  int t = blockIdx.x * blockDim.x + threadIdx.x;
  int e = t >> 4;
  if (e >= E) return;
  int c = (t & 15) * 4;
  float4 v = *(const float4*)(x + (size_t)src[e] * DIM + c);
  float* o = acc + (size_t)dst[e] * DIM + c;
  atomicAdd(o + 0, v.x);
  atomicAdd(o + 1, v.y);
  atomicAdd(o + 2, v.z);
  atomicAdd(o + 3, v.w);
}

// xnext[row] /= max(cnt[row],1);  xsum[row] += xnext[row]
__global__ void norm_accum_kernel(float* __restrict__ xnext,
                                  const float* __restrict__ cnt,
                                  float* __restrict__ xsum, int n) {
  int t = blockIdx.x * blockDim.x + threadIdx.x;
  int row = t >> 4;
  if (row >= n) return;
  int c = (t & 15) * 4;
  float inv = 1.0f / fmaxf(cnt[row], 1.0f);
  size_t off = (size_t)row * DIM + c;
  float4 v = *(float4*)(xnext + off);
  v.x *= inv; v.y *= inv; v.z *= inv; v.w *= inv;
  *(float4*)(xnext + off) = v;
  float4 s = *(float4*)(xsum + off);
  s.x += v.x; s.y += v.y; s.z += v.z; s.w += v.w;
  *(float4*)(xsum + off) = s;
}

// x[row] /= max(cnt[row],1)
__global__ void norm_rows_kernel(float* __restrict__ x,
                                 const float* __restrict__ cnt, int n) {
  int t = blockIdx.x * blockDim.x + threadIdx.x;
  int row = t >> 4;
  if (row >= n) return;
  int c = (t & 15) * 4;
  float inv = 1.0f / fmaxf(cnt[row], 1.0f);
  size_t off = (size_t)row * DIM + c;
  float4 v = *(float4*)(x + off);
  v.x *= inv; v.y *= inv; v.z *= inv; v.w *= inv;
  *(float4*)(x + off) = v;
}

__global__ void scale_kernel(const float* __restrict__ in, float* __restrict__ outp,
                             float s, int n) {
  int i = blockIdx.x * blockDim.x + threadIdx.x;
  if (i < n) outp[i] = in[i] * s;
}

// ---------------- WMMA logits kernel ----------------
// a[r] = ((rev[r,:] @ w_o + b_o) @ att_w) + att_b, one wave per 16-row tile.
// Uses V_WMMA_F32_16X16X4_F32: A is 16x4 (lanes 0-15: K0/K1, lanes 16-31: K2/K3),
// B is 4x16 (VGPR0: K0 low half / K2 high half; VGPR1: K1 / K3),
// C/D 16x16: VGPR v = row v (lanes 0-15) and row v+8 (lanes 16-31), lane&15 = column.
__global__ __launch_bounds__(32) void logits_wmma_kernel(
    const float* __restrict__ rev, const float* __restrict__ w_o,
    const float* __restrict__ b_o, const float* __restrict__ att_w,
    const float* __restrict__ att_b, float* __restrict__ a, int R) {
  int row0 = blockIdx.x * 16;
  int lane = threadIdx.x;
  int m = lane & 15;   // row of A tile / column of C tile
  int hi = lane >> 4;  // K half selector
  const float* arow = rev + (size_t)(row0 + m) * DIM;

  float rsum[8];
#pragma unroll
  for (int i = 0; i < 8; ++i) rsum[i] = 0.0f;

  for (int nt = 0; nt < 4; ++nt) {
    int n0 = nt * 16;
    v8f acc = {};
#pragma unroll
    for (int k = 0; k < 16; ++k) {
      int kb = k * 4 + hi * 2;
      v2f A, B;
      A.x = arow[kb];
      A.y = arow[kb + 1];
      B.x = w_o[(size_t)kb * DIM + n0 + m];
      B.y = w_o[(size_t)(kb + 1) * DIM + n0 + m];
      acc = __builtin_amdgcn_wmma_f32_16x16x4_f32(
          false, A, false, B, (short)0, acc, false, false);
    }
    int n = n0 + m;  // column this lane holds in C/D
    float aw = att_w[n];
    float bo = b_o[n];
#pragma unroll
    for (int i = 0; i < 8; ++i) rsum[i] += (acc[i] + bo) * aw;
  }

  // reduce over the 16 lanes of each half (xor offsets < 16 stay within half)
#pragma unroll
  for (int i = 0; i < 8; ++i) {
    float s = rsum[i];
    s += __shfl_xor(s, 1, 32);
    s += __shfl_xor(s, 2, 32);
    s += __shfl_xor(s, 4, 32);
    s += __shfl_xor(s, 8, 32);
    rsum[i] = s;
  }
  if (m == 0) {
    int rbase = row0 + hi * 8;
    float ab = att_b[0];
#pragma unroll
    for (int i = 0; i < 8; ++i) {
      int r = rbase + i;
      if (r < R) a[r] = rsum[i] + ab;
    }
  }
}

// ---------------- edge softmax kernels ----------------

__device__ inline void atomicMaxF(float* addr, float val) {
  unsigned int* ua = (unsigned int*)addr;
  unsigned int old = *ua;
  while (__uint_as_float(old) < val) {
    unsigned int assumed = old;
    old = atomicCAS(ua, assumed, __float_as_uint(val));
    if (old == assumed) break;
  }
}

__global__ void edge_max_kernel(const float* __restrict__ a,
                                const int* __restrict__ src,
                                const int* __restrict__ dst,
                                float* __restrict__ mx, int E) {
  int e = blockIdx.x * blockDim.x + threadIdx.x;
  if (e < E) atomicMaxF(mx + dst[e], a[src[e]]);
}

__global__ void edge_expden_kernel(const float* __restrict__ a,
                                   const int* __restrict__ src,
                                   const int* __restrict__ dst,
                                   const float* __restrict__ mx,
                                   float* __restrict__ den, int E) {
  int e = blockIdx.x * blockDim.x + threadIdx.x;
  if (e < E) atomicAdd(den + dst[e], expf(a[src[e]] - mx[dst[e]]));
}

// out[dst] += (exp(a[src]-mx[dst]) / max(den[dst],1e-9)) * rev[src]
__global__ void edge_weighted_scatter_kernel(
    const float* __restrict__ a, const float* __restrict__ rev,
    const int* __restrict__ src, const int* __restrict__ dst,
    const float* __restrict__ mx, const float* __restrict__ den,
    float* __restrict__ out, int E) {
  int t = blockIdx.x * blockDim.x + threadIdx.x;
  int e = t >> 4;
  if (e >= E) return;
  int c = (t & 15) * 4;
  int s = src[e], d = dst[e];
  float w = expf(a[s] - mx[d]) / fmaxf(den[d], 1e-9f);
  float4 v = *(const float4*)(rev + (size_t)s * DIM + c);
  float* o = out + (size_t)d * DIM + c;
  atomicAdd(o + 0, w * v.x);
  atomicAdd(o + 1, w * v.y);
  atomicAdd(o + 2, w * v.z);
  atomicAdd(o + 3, w * v.w);
}

// ---------------- host launch ----------------

static inline int nblk(long n, int b) { return (int)((n + b - 1) / b); }

extern "C" void kernel_launch(void* const* d_in, const int* in_sizes, int n_in,
                              void* d_out, int out_size, void* d_ws, size_t ws_size,
                              hipStream_t stream) {
  const float* emb    = (const float*)d_in[0];
  const float* w_o    = (const float*)d_in[1];
  const float* b_o    = (const float*)d_in[2];
  const float* att_w  = (const float*)d_in[3];
  const float* att_b  = (const float*)d_in[4];
  const int* node_ids = (const int*)d_in[5];
  const int* e1_src   = (const int*)d_in[6];
  const int* e1_dst   = (const int*)d_in[7];
  const int* e2_src   = (const int*)d_in[8];
  const int* e2_dst   = (const int*)d_in[9];
  const int* e3_src   = (const int*)d_in[10];
  const int* e3_dst   = (const int*)d_in[11];
  float* out = (float*)d_out;

  const int N  = in_sizes[5];
  const int E1 = in_sizes[6];
  const int E2 = in_sizes[8];
  const int E3 = in_sizes[10];
  const int M  = out_size / DIM;
  const int R  = 400000;  // review node count (fixed by reference model)

  float* ws = (float*)d_ws;
  float* cnt1 = ws; ws += N;
  float* xcur = ws; ws += (size_t)N * DIM;
  float* xnext= ws; ws += (size_t)N * DIM;
  float* xsum = ws; ws += (size_t)N * DIM;
  float* rev  = ws; ws += (size_t)R * DIM;
  float* cnt2 = ws; ws += R;
  float* alog = ws; ws += R;
  float* mx   = ws; ws += M;
  float* den  = ws; ws += M;

  const int B = 256;

  // 1) embedding gather (seeds xcur and the layer-sum accumulator)
  gather_rows_kernel<<<nblk((long)N * 16, B), B, 0, stream>>>(emb, node_ids, xcur, xsum, N);

  // 2) degree counts for e1 (same for all 3 layers)
  fill_kernel<<<nblk(N, B), B, 0, stream>>>(cnt1, 0.0f, N);
  count_dst_kernel<<<nblk(E1, B), B, 0, stream>>>(e1_dst, cnt1, E1);

  // 3) three LightGCN mean-propagation layers
  for (int l = 0; l < 3; ++l) {
    fill_kernel<<<nblk((long)N * DIM, B), B, 0, stream>>>(xnext, 0.0f, N * DIM);
    scatter_rows_kernel<<<nblk((long)E1 * 16, B), B, 0, stream>>>(xcur, e1_src, e1_dst, xnext, E1);
    norm_accum_kernel<<<nblk((long)N * 16, B), B, 0, stream>>>(xnext, cnt1, xsum, N);
    float* tmp = xcur; xcur = xnext; xnext = tmp;
  }

  // 4) readout: mean over 4 layer snapshots -> xnext (reused buffer)
  scale_kernel<<<nblk((long)N * DIM, B), B, 0, stream>>>(xsum, xnext, 0.25f, N * DIM);

  // 5) review mean-aggregation
  fill_kernel<<<nblk((long)R * DIM, B), B, 0, stream>>>(rev, 0.0f, R * DIM);
  fill_kernel<<<nblk(R, B), B, 0, stream>>>(cnt2, 0.0f, R);
  count_dst_kernel<<<nblk(E2, B), B, 0, stream>>>(e2_dst, cnt2, E2);
  scatter_rows_kernel<<<nblk((long)E2 * 16, B), B, 0, stream>>>(xnext, e2_src, e2_dst, rev, E2);
  norm_rows_kernel<<<nblk((long)R * 16, B), B, 0, stream>>>(rev, cnt2, R);

  // 6) attention logits via f32 WMMA (one wave per 16 review rows)
  logits_wmma_kernel<<<(R + 15) / 16, 32, 0, stream>>>(rev, w_o, b_o, att_w, att_b, alog, R);

  // 7) edge softmax over e3 + weighted scatter into output
  fill_kernel<<<nblk(M, B), B, 0, stream>>>(mx, -INFINITY, M);
  edge_max_kernel<<<nblk(E3, B), B, 0, stream>>>(alog, e3_src, e3_dst, mx, E3);
  fill_kernel<<<nblk(M, B), B, 0, stream>>>(den, 0.0f, M);
  edge_expden_kernel<<<nblk(E3, B), B, 0, stream>>>(alog, e3_src, e3_dst, mx, den, E3);
  fill_kernel<<<nblk((long)M * DIM, B), B, 0, stream>>>(out, 0.0f, M * DIM);
  edge_weighted_scatter_kernel<<<nblk((long)E3 * 16, B), B, 0, stream>>>(
      alog, rev, e3_src, e3_dst, mx, den, out, E3);
}